// BayesNAM_76605036691471
// MI455X (gfx1250) — compile-verified
//
#include <hip/hip_runtime.h>
#include <hip/hip_bf16.h>

typedef float v2f __attribute__((ext_vector_type(2)));
typedef float v8f __attribute__((ext_vector_type(8)));

#define B_DIM 16384
#define F_DIM 64
#define H_DIM 64
#define NEG_SLOPE 0.01f
#define LOG_PS   (-2.302585092994046f)   // ln(0.1)
#define INV_2S2  50.0f                   // 1/(2*0.1^2)

// workspace layout (float offsets)
#define W1SW_OFF 0                 // 64 * 2048 float2 = 262144 floats (B-fragment swizzled W1eff)
#define S0_OFF   262144            // 64*64
#define B0_OFF   266240            // 64*64
#define B1_OFF   270336            // 64*64
#define W2_OFF   274432            // 64*64
#define B2_OFF   278528            // 64
#define FT_OFF   278592            // x transposed: 64 * 16384

__device__ __forceinline__ float lrelu(float x) { return fmaxf(x, NEG_SLOPE * x); }

__device__ __forceinline__ float klterm(float mu, float ls) {
    return LOG_PS - ls + (__expf(2.0f * ls) + mu * mu) * INV_2S2 - 0.5f;
}

// ---------------------------------------------------------------- init: y[b]=bias, kl=0
__global__ void init_kernel(const float* __restrict__ bias, float* __restrict__ out) {
    int t = blockIdx.x * blockDim.x + threadIdx.x;
    if (t < B_DIM) out[t] = bias[0];
    if (t == B_DIM) out[B_DIM] = 0.0f;
}

// ---------------------------------------------------------------- transpose x -> f-major
__global__ void transpose_kernel(const float* __restrict__ f, float* __restrict__ ws) {
    int t = blockIdx.x * blockDim.x + threadIdx.x;   // 0 .. B*F-1, coalesced read
    int b  = t >> 6;
    int fi = t & 63;
    ws[FT_OFF + fi * B_DIM + b] = f[t];
}

// ---------------------------------------------------------------- effective weights + B-fragment swizzle
// W1sw[f][kk][nt][lane] = { W1eff[4kk+2*(lane>=16)][16nt+(lane&15)],
//                           W1eff[4kk+2*(lane>=16)+1][16nt+(lane&15)] }
__global__ void prep_kernel(const float* __restrict__ wmu0, const float* __restrict__ wls0,
                            const float* __restrict__ bmu0, const float* __restrict__ bls0,
                            const float* __restrict__ ew0,  const float* __restrict__ eb0,
                            const float* __restrict__ wmu1, const float* __restrict__ wls1,
                            const float* __restrict__ bmu1, const float* __restrict__ bls1,
                            const float* __restrict__ ew1,  const float* __restrict__ eb1,
                            const float* __restrict__ wmu2, const float* __restrict__ wls2,
                            const float* __restrict__ bmu2, const float* __restrict__ bls2,
                            const float* __restrict__ ew2,  const float* __restrict__ eb2,
                            float* __restrict__ ws) {
    int t = blockIdx.x * blockDim.x + threadIdx.x;    // 0 .. 131071
    if (t < F_DIM * 2048) {
        int lane = t & 31;
        int nt   = (t >> 5) & 3;
        int kk   = (t >> 7) & 15;
        int fi   = t >> 11;
        int i0   = 4 * kk + 2 * (lane >> 4);
        int col  = 16 * nt + (lane & 15);
        int g0   = (fi * H_DIM + i0) * H_DIM + col;
        int g1   = g0 + H_DIM;                         // row i0+1
        float e0 = wmu1[g0] + __expf(wls1[g0]) * ew1[g0];
        float e1 = wmu1[g1] + __expf(wls1[g1]) * ew1[g1];
        float2* w1sw = (float2*)(ws + W1SW_OFF);
        w1sw[t] = make_float2(e0, e1);
    }
    if (t < F_DIM * H_DIM) {
        ws[S0_OFF + t] = wmu0[t] + __expf(wls0[t]) * ew0[t];
        ws[B0_OFF + t] = bmu0[t] + __expf(bls0[t]) * eb0[t];
        ws[B1_OFF + t] = bmu1[t] + __expf(bls1[t]) * eb1[t];
        ws[W2_OFF + t] = wmu2[t] + __expf(wls2[t]) * ew2[t];
    }
    if (t < F_DIM) {
        ws[B2_OFF + t] = bmu2[t] + __expf(bls2[t]) * eb2[t];
    }
}

// ---------------------------------------------------------------- KL reduction (weighted by per-mean counts)
__global__ void kl_kernel(const float* __restrict__ wmu0, const float* __restrict__ wls0,
                          const float* __restrict__ bmu0, const float* __restrict__ bls0,
                          const float* __restrict__ wmu1, const float* __restrict__ wls1,
                          const float* __restrict__ bmu1, const float* __restrict__ bls1,
                          const float* __restrict__ wmu2, const float* __restrict__ wls2,
                          const float* __restrict__ bmu2, const float* __restrict__ bls2,
                          float* __restrict__ out) {
    __shared__ float red[256];
    int tid    = threadIdx.x;
    int gid    = blockIdx.x * blockDim.x + tid;
    int stride = gridDim.x * blockDim.x;
    float local = 0.0f;
    for (int i = gid; i < F_DIM * H_DIM * H_DIM; i += stride)
        local += klterm(wmu1[i], wls1[i]) * (1.0f / 4096.0f);
    for (int i = gid; i < F_DIM * H_DIM; i += stride)
        local += (klterm(wmu0[i], wls0[i]) + klterm(bmu0[i], bls0[i]) +
                  klterm(bmu1[i], bls1[i]) + klterm(wmu2[i], wls2[i])) * (1.0f / 64.0f);
    for (int i = gid; i < F_DIM; i += stride)
        local += klterm(bmu2[i], bls2[i]);
    red[tid] = local;
    __syncthreads();
    for (int s = 128; s > 0; s >>= 1) {
        if (tid < s) red[tid] += red[tid + s];
        __syncthreads();
    }
    if (tid == 0) atomicAdd(out + B_DIM, red[0]);
}

// ---------------------------------------------------------------- fused MLP: grid (64 btiles, 64 f)
__global__ __launch_bounds__(256) void mlp_kernel(const float* __restrict__ ws,
                                                  float* __restrict__ y) {
    __shared__ float2 w1s[2048];                 // 16 KB swizzled W1eff[f]
    __shared__ float  s0s[64], b0s[64], b1s[64], w2s[64];

    const int fi = blockIdx.y;
    const int t  = threadIdx.x;

    const float2* w1g = (const float2*)(ws + W1SW_OFF) + fi * 2048;
#pragma unroll
    for (int j = 0; j < 8; ++j) w1s[t + j * 256] = w1g[t + j * 256];
    if (t < 64) {
        s0s[t] = ws[S0_OFF + fi * 64 + t];
        b0s[t] = ws[B0_OFF + fi * 64 + t];
        b1s[t] = ws[B1_OFF + fi * 64 + t];
        w2s[t] = ws[W2_OFF + fi * 64 + t];
    }
    __syncthreads();

    const float b2v  = ws[B2_OFF + fi];
    const int   lane = t & 31;
    const int   wave = t >> 5;
    const int   hi   = lane >> 4;      // half-wave selector (K offset / M offset)
    const int   m    = lane & 15;
    const float* fT  = ws + FT_OFF + fi * B_DIM;

#pragma unroll
    for (int sub = 0; sub < 2; ++sub) {
        const int rowbase = blockIdx.x * 256 + (wave * 2 + sub) * 16;
        const float x = fT[rowbase + m];          // layer-0 scalar input for row m

        v8f c[4] = {};                            // 16x64 fp32 accumulator tile
#pragma unroll
        for (int kk = 0; kk < 16; ++kk) {         // K = 64 in steps of 4
            const int i0 = 4 * kk + 2 * hi;
            // fused layer 0 -> A fragment (A[m][k] layout per ISA 7.12.2)
            v2f a;
            a[0] = lrelu(fmaf(x, s0s[i0],     b0s[i0]));
            a[1] = lrelu(fmaf(x, s0s[i0 + 1], b0s[i0 + 1]));
#pragma unroll
            for (int nt = 0; nt < 4; ++nt) {
                float2 bw = w1s[(kk * 4 + nt) * 32 + lane];   // conflict-free b64
                v2f b; b[0] = bw.x; b[1] = bw.y;
                c[nt] = __builtin_amdgcn_wmma_f32_16x16x4_f32(
                            false, a, false, b, (short)0, c[nt], false, false);
            }
        }

        // epilogue: +b1, lrelu, dot with w2 (D layout: vgpr r -> row r+8*hi, col 16nt+m)
        float rs[8];
#pragma unroll
        for (int r = 0; r < 8; ++r) rs[r] = 0.0f;
#pragma unroll
        for (int nt = 0; nt < 4; ++nt) {
            const int   col = nt * 16 + m;
            const float w2v = w2s[col];
            const float b1v = b1s[col];
#pragma unroll
            for (int r = 0; r < 8; ++r) {
                float h = lrelu(c[nt][r] + b1v);
                rs[r] = fmaf(h, w2v, rs[r]);
            }
        }
        // reduce over the 16 lanes of each half (cols) -> per-row sums
#pragma unroll
        for (int r = 0; r < 8; ++r) {
#pragma unroll
            for (int off = 1; off < 16; off <<= 1)
                rs[r] += __shfl_xor(rs[r], off, 32);
        }
        if (m < 8) {
            float myv = 0.0f;
#pragma unroll
            for (int r = 0; r < 8; ++r)
                if (m == r) myv = rs[r];
            atomicAdd(&y[rowbase + hi * 8 + m], myv + b2v);
        }
    }
}

extern "C" void kernel_launch(void* const* d_in, const int* in_sizes, int n_in,
                              void* d_out, int out_size, void* d_ws, size_t ws_size,
                              hipStream_t stream) {
    (void)in_sizes; (void)n_in; (void)out_size; (void)ws_size;
    const float* f    = (const float*)d_in[0];
    const float* wmu0 = (const float*)d_in[1];
    const float* wls0 = (const float*)d_in[2];
    const float* bmu0 = (const float*)d_in[3];
    const float* bls0 = (const float*)d_in[4];
    const float* ew0  = (const float*)d_in[5];
    const float* eb0  = (const float*)d_in[6];
    const float* wmu1 = (const float*)d_in[7];
    const float* wls1 = (const float*)d_in[8];
    const float* bmu1 = (const float*)d_in[9];
    const float* bls1 = (const float*)d_in[10];
    const float* ew1  = (const float*)d_in[11];
    const float* eb1  = (const float*)d_in[12];
    const float* wmu2 = (const float*)d_in[13];
    const float* wls2 = (const float*)d_in[14];
    const float* bmu2 = (const float*)d_in[15];
    const float* bls2 = (const float*)d_in[16];
    const float* ew2  = (const float*)d_in[17];
    const float* eb2  = (const float*)d_in[18];
    const float* bias = (const float*)d_in[19];

    float* out = (float*)d_out;   // [0..16383] = y, [16384] = kl_total
    float* ws  = (float*)d_ws;

    init_kernel<<<(B_DIM + 256) / 256, 256, 0, stream>>>(bias, out);
    transpose_kernel<<<(B_DIM * F_DIM) / 256, 256, 0, stream>>>(f, ws);
    prep_kernel<<<(F_DIM * 2048) / 256, 256, 0, stream>>>(
        wmu0, wls0, bmu0, bls0, ew0, eb0,
        wmu1, wls1, bmu1, bls1, ew1, eb1,
        wmu2, wls2, bmu2, bls2, ew2, eb2, ws);
    kl_kernel<<<256, 256, 0, stream>>>(
        wmu0, wls0, bmu0, bls0, wmu1, wls1, bmu1, bls1, wmu2, wls2, bmu2, bls2, out);
    mlp_kernel<<<dim3(B_DIM / 256, F_DIM), 256, 0, stream>>>(ws, out);
}